// HAN_7267084664969
// MI455X (gfx1250) — compile-verified
//
#include <hip/hip_runtime.h>

// ---------------- problem constants (match reference) ----------------
#define HH       4
#define DD       64
#define CC       256
#define NNEWS    50000
#define NUSER    100000
#define FIN      512
#define EPOSTS   1000000
#define ESIM     500000
#define NEGSLOPE 0.2f

typedef __attribute__((ext_vector_type(2))) float v2f;
typedef __attribute__((ext_vector_type(8))) float v8f;

__device__ __forceinline__ float lrelu(float x) { return x > 0.f ? x : NEGSLOPE * x; }

// =====================================================================
// WMMA f32 GEMM:  C[M,256] = A[M,K] * B[K,256] (+bias)
// MODE 0: store  C = acc + bias                         (projection)
// MODE 1: relu(A) on load; colsum += tanh(acc + bias)   (semantic attn)
// Block: 256 thr = 8 waves; tile 64(M) x 256(N); K chunk 32; software
// pipelined: next chunk's global loads issued before current compute.
// =====================================================================
template <int MODE>
__global__ __launch_bounds__(256) void gemm_wmma(
    const float* __restrict__ A, const float* __restrict__ B,
    const float* __restrict__ bias, float* __restrict__ Cout, int M, int K)
{
    __shared__ float As[64 * 40];        // 64 rows x 32 K, padded to 40
    __shared__ float Bs[16 * 256 * 2];   // 16 K-pairs x 256 N x float2

    const int tid  = threadIdx.x;
    const int wave = tid >> 5, lane = tid & 31;
    const int mw = wave & 3, nw = wave >> 2;
    const int lhi = lane >> 4, ln = lane & 15;
    const int blockRow = blockIdx.x * 64;

    v8f acc[8];
#pragma unroll
    for (int t = 0; t < 8; ++t) acc[t] = (v8f){0.f,0.f,0.f,0.f,0.f,0.f,0.f,0.f};

    float4 ra[2], rb[8];
    // prefetch chunk 0
#pragma unroll
    for (int i = 0; i < 2; ++i) {
        int v = tid + i * 256;
        int row = v >> 3, col = (v & 7) * 4;
        int grow = blockRow + row;
        float4 f = make_float4(0.f, 0.f, 0.f, 0.f);
        if (grow < M) f = *(const float4*)&A[(size_t)grow * K + col];
        ra[i] = f;
    }
#pragma unroll
    for (int i = 0; i < 8; ++i) {
        int v = i * 256 + tid;
        int kr = v >> 6, col = (v & 63) * 4;
        rb[i] = *(const float4*)&B[(size_t)kr * CC + col];
    }

    for (int k0 = 0; k0 < K; k0 += 32) {
        // ---- write staged registers to LDS ----
#pragma unroll
        for (int i = 0; i < 2; ++i) {
            int v = tid + i * 256;
            int row = v >> 3, col = (v & 7) * 4;
            float4 f = ra[i];
            if (MODE == 1) { f.x = fmaxf(f.x, 0.f); f.y = fmaxf(f.y, 0.f);
                             f.z = fmaxf(f.z, 0.f); f.w = fmaxf(f.w, 0.f); }
            float* p = &As[row * 40 + col];
            p[0] = f.x; p[1] = f.y; p[2] = f.z; p[3] = f.w;
        }
#pragma unroll
        for (int i = 0; i < 8; ++i) {
            int v = i * 256 + tid;
            int kr = v >> 6, col = (v & 63) * 4;
            float4 f = rb[i];
            float* p = &Bs[(((kr >> 1) * 256 + col) << 1) + (kr & 1)];
            p[0] = f.x; p[2] = f.y; p[4] = f.z; p[6] = f.w;
        }
        __syncthreads();

        // ---- prefetch next chunk while computing ----
        int kn = k0 + 32;
        if (kn < K) {
#pragma unroll
            for (int i = 0; i < 2; ++i) {
                int v = tid + i * 256;
                int row = v >> 3, col = (v & 7) * 4;
                int grow = blockRow + row;
                float4 f = make_float4(0.f, 0.f, 0.f, 0.f);
                if (grow < M) f = *(const float4*)&A[(size_t)grow * K + kn + col];
                ra[i] = f;
            }
#pragma unroll
            for (int i = 0; i < 8; ++i) {
                int v = i * 256 + tid;
                int kr = v >> 6, col = (v & 63) * 4;
                rb[i] = *(const float4*)&B[(size_t)(kn + kr) * CC + col];
            }
        }

        // ---- WMMA over the chunk ----
#pragma unroll
        for (int kk = 0; kk < 32; kk += 4) {
            // A frag: lane L holds M=L%16, K = kk + (L<16 ? 0,1 : 2,3)
            const float* ap = &As[(mw * 16 + ln) * 40 + kk + 2 * lhi];
            v2f af; af[0] = ap[0]; af[1] = ap[1];
            int bp = (kk >> 1) + lhi;
            v2f bf[8];
#pragma unroll
            for (int t = 0; t < 8; ++t) {
                const float* bq = &Bs[((bp * 256 + nw * 128 + t * 16 + ln) << 1)];
                bf[t][0] = bq[0]; bf[t][1] = bq[1];
            }
#pragma unroll
            for (int t = 0; t < 8; ++t)
                acc[t] = __builtin_amdgcn_wmma_f32_16x16x4_f32(
                    false, af, false, bf[t], (short)0, acc[t], false, false);
        }
        __syncthreads();
    }

    if (MODE == 0) {
#pragma unroll
        for (int t = 0; t < 8; ++t) {
            int col = nw * 128 + t * 16 + ln;
            float bv = bias[col];
#pragma unroll
            for (int r = 0; r < 8; ++r) {
                int row = blockRow + mw * 16 + r + 8 * lhi;
                if (row < M) Cout[(size_t)row * CC + col] = acc[t][r] + bv;
            }
        }
    } else {
#pragma unroll
        for (int t = 0; t < 8; ++t) {
            int col = nw * 128 + t * 16 + ln;
            float bv = bias[col];
            float s = 0.f;
#pragma unroll
            for (int r = 0; r < 8; ++r) {
                int row = blockRow + mw * 16 + r + 8 * lhi;
                float v = tanhf(acc[t][r] + bv);
                s += (row < M) ? v : 0.f;
            }
            s += __shfl_down(s, 16, 32);
            if (lhi == 0) atomicAdd(&Cout[col], s);
        }
    }
}

// ---------------- al[n][h] = sum_d h[n,h,d]*att[h,d]  (wave per node) ---
__global__ __launch_bounds__(256) void al_kernel(
    const float* __restrict__ h, const float* __restrict__ att,
    float* __restrict__ al, int N)
{
    int node = (int)((blockIdx.x * blockDim.x + threadIdx.x) >> 5);
    int lane = threadIdx.x & 31;
    if (node >= N) return;
    int c0 = lane * 8;
    const float4* hp = (const float4*)&h[(size_t)node * CC + c0];
    const float4* ap = (const float4*)&att[c0];
    float4 h0 = hp[0], h1 = hp[1], a0 = ap[0], a1 = ap[1];
    float s = h0.x*a0.x + h0.y*a0.y + h0.z*a0.z + h0.w*a0.w
            + h1.x*a1.x + h1.y*a1.y + h1.z*a1.z + h1.w*a1.w;
    s += __shfl_xor(s, 1, 32);
    s += __shfl_xor(s, 2, 32);
    s += __shfl_xor(s, 4, 32);
    if ((lane & 7) == 0) al[(size_t)node * HH + (lane >> 3)] = s;
}

// ---------------- CSR construction -----------------------------------
__global__ __launch_bounds__(256) void deg_kernel(
    const int* __restrict__ dst, int* __restrict__ deg, int E)
{
    int e = blockIdx.x * blockDim.x + threadIdx.x;
    if (e < E) atomicAdd(&deg[dst[e]], 1);
}

// single-block exclusive scan; also initializes the scatter cursors
__global__ __launch_bounds__(256) void scan_kernel(
    const int* __restrict__ deg, int* __restrict__ off,
    int* __restrict__ cursor, int N)
{
    __shared__ int buf[256];
    __shared__ int carry;
    int tid = threadIdx.x;
    if (tid == 0) carry = 0;
    __syncthreads();
    for (int base = 0; base < N; base += 256) {
        int v = (base + tid < N) ? deg[base + tid] : 0;
        buf[tid] = v;
        __syncthreads();
        for (int s = 1; s < 256; s <<= 1) {
            int t = (tid >= s) ? buf[tid - s] : 0;
            __syncthreads();
            buf[tid] += t;
            __syncthreads();
        }
        int total = buf[255];
        int excl  = buf[tid] - v;
        int c     = carry;
        __syncthreads();
        if (tid == 0) carry = c + total;
        if (base + tid < N) {
            off[base + tid]    = c + excl;
            cursor[base + tid] = c + excl;
        }
        __syncthreads();
    }
    if (tid == 0) off[N] = carry;
}

__global__ __launch_bounds__(256) void scatter_kernel(
    const int* __restrict__ src, const int* __restrict__ dst,
    int* __restrict__ cursor, int* __restrict__ elist, int E)
{
    int e = blockIdx.x * blockDim.x + threadIdx.x;
    if (e >= E) return;
    int pos = atomicAdd(&cursor[dst[e]], 1);
    elist[pos] = src[e];
}

// ---------------- CSR gather aggregation: wave per dst node -----------
// lane covers 8 channels of one head; softmax stats live in registers.
__global__ __launch_bounds__(256) void agg_csr_kernel(
    const int* __restrict__ off, const int* __restrict__ elist,
    const float* __restrict__ als, const float* __restrict__ ald,
    const float* __restrict__ hsrc, float* __restrict__ out, int N)
{
    int d = (int)((blockIdx.x * blockDim.x + threadIdx.x) >> 5);
    int lane = threadIdx.x & 31;
    if (d >= N) return;
    int h  = lane >> 3;
    int c0 = lane * 8;
    int beg = off[d], end = off[d + 1];
    float aldv = ald[(size_t)d * HH + h];

    // pass 1: per-head max
    float mx = -1e30f;
    for (int i = beg; i < end; ++i) {
        int s = elist[i];
        mx = fmaxf(mx, lrelu(als[(size_t)s * HH + h] + aldv));
    }
    // pass 2: per-head exp-sum
    float z = 0.f;
    for (int i = beg; i < end; ++i) {
        int s = elist[i];
        z += expf(lrelu(als[(size_t)s * HH + h] + aldv) - mx);
    }
    float zinv = 1.f / (z + 1e-16f);
    // pass 3: weighted accumulate (registers only; no atomics)
    float acc[8] = {0.f,0.f,0.f,0.f,0.f,0.f,0.f,0.f};
    for (int i = beg; i < end; ++i) {
        int s = elist[i];
        float w = expf(lrelu(als[(size_t)s * HH + h] + aldv) - mx) * zinv;
        const float4* hp = (const float4*)&hsrc[(size_t)s * CC + c0];
        float4 a = hp[0], b = hp[1];
        acc[0] += a.x * w; acc[1] += a.y * w; acc[2] += a.z * w; acc[3] += a.w * w;
        acc[4] += b.x * w; acc[5] += b.y * w; acc[6] += b.z * w; acc[7] += b.w * w;
    }
    float4* op = (float4*)&out[(size_t)d * CC + c0];
    op[0] = make_float4(acc[0], acc[1], acc[2], acc[3]);
    op[1] = make_float4(acc[4], acc[5], acc[6], acc[7]);
}

// ---------------- semantic softmax over the 2 metapaths ---------------
__global__ void score_kernel(const float* __restrict__ colsum,
                             const float* __restrict__ q, float* __restrict__ attn)
{
    int lane = threadIdx.x & 31;
    float s0 = 0.f, s1 = 0.f;
    for (int c = lane; c < CC; c += 32) {
        s0 += q[c] * colsum[c];
        s1 += q[c] * colsum[CC + c];
    }
    for (int msk = 16; msk; msk >>= 1) {
        s0 += __shfl_xor(s0, msk, 32);
        s1 += __shfl_xor(s1, msk, 32);
    }
    if (lane == 0) {
        s0 /= (float)NNEWS; s1 /= (float)NNEWS;
        float mx = fmaxf(s0, s1);
        float e0 = expf(s0 - mx), e1 = expf(s1 - mx);
        attn[0] = e0 / (e0 + e1);
        attn[1] = e1 / (e0 + e1);
    }
}

// ---------------- fuse + classifier head (wave per node) --------------
__global__ __launch_bounds__(256) void final_kernel(
    const float* __restrict__ outp, const float* __restrict__ outs,
    const float* __restrict__ attn, const float* __restrict__ lw,
    const float* __restrict__ lb, float* __restrict__ y, int N)
{
    int n = (int)((blockIdx.x * blockDim.x + threadIdx.x) >> 5);
    int lane = threadIdx.x & 31;
    if (n >= N) return;
    float a0 = attn[0], a1 = attn[1];
    int c0 = lane * 8;
    const float4* pp = (const float4*)&outp[(size_t)n * CC + c0];
    const float4* sp = (const float4*)&outs[(size_t)n * CC + c0];
    float4 p0 = pp[0], p1 = pp[1], q0 = sp[0], q1 = sp[1];
    float f[8] = {
        a0 * fmaxf(p0.x, 0.f) + a1 * fmaxf(q0.x, 0.f),
        a0 * fmaxf(p0.y, 0.f) + a1 * fmaxf(q0.y, 0.f),
        a0 * fmaxf(p0.z, 0.f) + a1 * fmaxf(q0.z, 0.f),
        a0 * fmaxf(p0.w, 0.f) + a1 * fmaxf(q0.w, 0.f),
        a0 * fmaxf(p1.x, 0.f) + a1 * fmaxf(q1.x, 0.f),
        a0 * fmaxf(p1.y, 0.f) + a1 * fmaxf(q1.y, 0.f),
        a0 * fmaxf(p1.z, 0.f) + a1 * fmaxf(q1.z, 0.f),
        a0 * fmaxf(p1.w, 0.f) + a1 * fmaxf(q1.w, 0.f)};
    float y0 = 0.f, y1 = 0.f;
#pragma unroll
    for (int j = 0; j < 8; ++j) {
        y0 += f[j] * lw[(size_t)(c0 + j) * 2 + 0];
        y1 += f[j] * lw[(size_t)(c0 + j) * 2 + 1];
    }
    for (int msk = 16; msk; msk >>= 1) {
        y0 += __shfl_xor(y0, msk, 32);
        y1 += __shfl_xor(y1, msk, 32);
    }
    if (lane == 0) {
        y[(size_t)n * 2 + 0] = y0 + lb[0];
        y[(size_t)n * 2 + 1] = y1 + lb[1];
    }
}

__global__ void fill_kernel(float* __restrict__ p, size_t n, float v) {
    size_t i = (size_t)blockIdx.x * blockDim.x + threadIdx.x;
    if (i < n) p[i] = v;
}

// =====================================================================
extern "C" void kernel_launch(void* const* d_in, const int* in_sizes, int n_in,
                              void* d_out, int out_size, void* d_ws, size_t ws_size,
                              hipStream_t stream)
{
    (void)in_sizes; (void)n_in; (void)out_size; (void)ws_size;
    const float* x_news   = (const float*)d_in[0];
    const float* x_user   = (const float*)d_in[1];
    const int*   psrc     = (const int*)d_in[2];
    const int*   pdst     = (const int*)d_in[3];
    const int*   ssrc     = (const int*)d_in[4];
    const int*   sdst     = (const int*)d_in[5];
    const float* pw_news  = (const float*)d_in[6];
    const float* pb_news  = (const float*)d_in[7];
    const float* pw_user  = (const float*)d_in[8];
    const float* pb_user  = (const float*)d_in[9];
    const float* att_sp   = (const float*)d_in[10];
    const float* att_dp   = (const float*)d_in[11];
    const float* att_ss   = (const float*)d_in[12];
    const float* att_ds   = (const float*)d_in[13];
    const float* qv       = (const float*)d_in[14];
    const float* klw      = (const float*)d_in[15];
    const float* klb      = (const float*)d_in[16];
    const float* lw       = (const float*)d_in[17];
    const float* lb       = (const float*)d_in[18];

    float* ws = (float*)d_ws;
    size_t off = 0;
    float* h_news = ws + off; off += (size_t)NNEWS * CC;
    float* h_user = ws + off; off += (size_t)NUSER * CC;
    float* out_p  = ws + off; off += (size_t)NNEWS * CC;
    float* out_s  = ws + off; off += (size_t)NNEWS * CC;
    float* al_sp  = ws + off; off += (size_t)NUSER * HH;
    float* al_dp  = ws + off; off += (size_t)NNEWS * HH;
    float* al_ss  = ws + off; off += (size_t)NNEWS * HH;
    float* al_ds  = ws + off; off += (size_t)NNEWS * HH;
    float* colsum = ws + off; off += 2 * CC;
    float* attn   = ws + off; off += 2;
    // integer CSR storage (4B elements, aliased onto float workspace)
    int* ideg_p   = (int*)(ws + off); off += NNEWS;       // degree, then cursor
    int* ioff_p   = (int*)(ws + off); off += NNEWS + 1;
    int* ielist_p = (int*)(ws + off); off += EPOSTS;
    int* ideg_s   = (int*)(ws + off); off += NNEWS;
    int* ioff_s   = (int*)(ws + off); off += NNEWS + 1;
    int* ielist_s = (int*)(ws + off); off += ESIM;

    // init (colsum + attn, degree counters)
    fill_kernel<<<2, 256, 0, stream>>>(colsum, 2 * CC + 2, 0.f);
    fill_kernel<<<(NNEWS + 255) / 256, 256, 0, stream>>>((float*)ideg_p, NNEWS, 0.f);
    fill_kernel<<<(NNEWS + 255) / 256, 256, 0, stream>>>((float*)ideg_s, NNEWS, 0.f);

    // node projections (f32 WMMA, exact precision)
    gemm_wmma<0><<<dim3((NNEWS + 63) / 64), 256, 0, stream>>>(x_news, pw_news, pb_news, h_news, NNEWS, FIN);
    gemm_wmma<0><<<dim3((NUSER + 63) / 64), 256, 0, stream>>>(x_user, pw_user, pb_user, h_user, NUSER, FIN);

    // per-node attention logits
    al_kernel<<<(NUSER + 7) / 8, 256, 0, stream>>>(h_user, att_sp, al_sp, NUSER);
    al_kernel<<<(NNEWS + 7) / 8, 256, 0, stream>>>(h_news, att_dp, al_dp, NNEWS);
    al_kernel<<<(NNEWS + 7) / 8, 256, 0, stream>>>(h_news, att_ss, al_ss, NNEWS);
    al_kernel<<<(NNEWS + 7) / 8, 256, 0, stream>>>(h_news, att_ds, al_ds, NNEWS);

    // build CSR for both edge types (counts -> scan -> scatter)
    deg_kernel<<<(EPOSTS + 255) / 256, 256, 0, stream>>>(pdst, ideg_p, EPOSTS);
    deg_kernel<<<(ESIM + 255) / 256, 256, 0, stream>>>(sdst, ideg_s, ESIM);
    scan_kernel<<<1, 256, 0, stream>>>(ideg_p, ioff_p, ideg_p, NNEWS);   // cursor aliases deg
    scan_kernel<<<1, 256, 0, stream>>>(ideg_s, ioff_s, ideg_s, NNEWS);
    scatter_kernel<<<(EPOSTS + 255) / 256, 256, 0, stream>>>(psrc, pdst, ideg_p, ielist_p, EPOSTS);
    scatter_kernel<<<(ESIM + 255) / 256, 256, 0, stream>>>(ssrc, sdst, ideg_s, ielist_s, ESIM);

    // gather-based edge softmax + aggregation (no float atomics)
    agg_csr_kernel<<<(NNEWS + 7) / 8, 256, 0, stream>>>(ioff_p, ielist_p, al_sp, al_dp, h_user, out_p, NNEWS);
    agg_csr_kernel<<<(NNEWS + 7) / 8, 256, 0, stream>>>(ioff_s, ielist_s, al_ss, al_ds, h_news, out_s, NNEWS);

    // semantic attention: colsum_p = sum_n tanh(relu(out)@klw + klb)
    gemm_wmma<1><<<dim3((NNEWS + 63) / 64), 256, 0, stream>>>(out_p, klw, klb, colsum, NNEWS, CC);
    gemm_wmma<1><<<dim3((NNEWS + 63) / 64), 256, 0, stream>>>(out_s, klw, klb, colsum + CC, NNEWS, CC);
    score_kernel<<<1, 32, 0, stream>>>(colsum, qv, attn);

    // fuse + classifier head
    final_kernel<<<(NNEWS + 7) / 8, 256, 0, stream>>>(out_p, out_s, attn, lw, lb, (float*)d_out, NNEWS);
}